// HGAT_20246475833866
// MI455X (gfx1250) — compile-verified
//
#include <hip/hip_runtime.h>
#include <cmath>
#include <cstddef>

#define N_NODES 20000
#define M_HYPER 2000
#define E_INC   320000
// feature width is 128 everywhere (D_MODEL == HID == 128)

typedef __attribute__((ext_vector_type(2))) float v2f;
typedef __attribute__((ext_vector_type(8))) float v8f;

__device__ __forceinline__ float leaky02(float v) { return v > 0.f ? v : 0.2f * v; }

__device__ __forceinline__ void atomic_max_f32(float* addr, float val) {
  // IEEE-754 monotone bit trick; buffer initialized to -inf (0xff800000)
  if (val >= 0.f)
    atomicMax(reinterpret_cast<int*>(addr), __float_as_int(val));
  else
    atomicMin(reinterpret_cast<unsigned int*>(addr), __float_as_uint(val));
}

// ---------------------------------------------------------------------------
// WMMA f32 GEMM: C[M,N] = act(A[M,K] @ B[K,N] + bias)
// One wave per 16x16 tile, 4 waves per block along N.
// Fragment layout per CDNA5 ISA 7.12.2 (V_WMMA_F32_16X16X4_F32):
//   A 16x4 : lanes 0-15 row M=l16 {K=kk,kk+1}; lanes 16-31 same rows {K=kk+2,kk+3}
//   B 4x16 : lanes 0-15 col N=l16 {K=kk,kk+1}; lanes 16-31 {K=kk+2,kk+3}
//   C 16x16: VGPR i -> row i (lanes 0-15) / row i+8 (lanes 16-31), col = l16
// ---------------------------------------------------------------------------
__global__ void wmma_gemm_f32(const float* __restrict__ A, const float* __restrict__ B,
                              const float* __restrict__ bias, float* __restrict__ C,
                              int Mdim, int Ndim, int Kdim, int act)
{
  const int lane = threadIdx.x & 31;
  const int wave = threadIdx.x >> 5;
  const int tm = blockIdx.x;
  const int tn = blockIdx.y * 4 + wave;
  if ((tm << 4) >= Mdim || (tn << 4) >= Ndim) return;   // wave-uniform guard
  const int half = lane >> 4;
  const int l16  = lane & 15;
  const float* Arow = A + (size_t)((tm << 4) + l16) * Kdim;
  const float* Bcol = B + (tn << 4) + l16;
  v8f acc = {0.f, 0.f, 0.f, 0.f, 0.f, 0.f, 0.f, 0.f};
  for (int kk = 0; kk < Kdim; kk += 4) {
    const int k0 = kk + (half << 1);
    v2f a = { Arow[k0], Arow[k0 + 1] };
    v2f b = { Bcol[(size_t)k0 * Ndim], Bcol[(size_t)(k0 + 1) * Ndim] };
    acc = __builtin_amdgcn_wmma_f32_16x16x4_f32(false, a, false, b, (short)0, acc,
                                                false, false);
  }
  const int col = (tn << 4) + l16;
  const float bv = bias ? bias[col] : 0.f;
#pragma unroll
  for (int i = 0; i < 8; ++i) {
    const int row = (tm << 4) + (half << 3) + i;
    float v = acc[i] + bv;
    if (act) v = leaky02(v);
    C[(size_t)row * Ndim + col] = v;
  }
}

// ---------------------------------------------------------------------------
// BatchNorm (training-mode batch stats, biased variance), C = 128 columns
// ---------------------------------------------------------------------------
__global__ void bn_stats(const float* __restrict__ x, const float* __restrict__ resid,
                         float* __restrict__ sums /* [256]: sum, sumsq */, int rows)
{
  const int c = threadIdx.x;                 // 128 threads = 1 column each
  const int per = (rows + gridDim.x - 1) / gridDim.x;
  const int r0 = blockIdx.x * per;
  const int r1 = min(rows, r0 + per);
  float s = 0.f, q = 0.f;
  for (int r = r0; r < r1; ++r) {
    float v = x[(size_t)r * 128 + c];
    if (resid) v += resid[(size_t)r * 128 + c];
    s += v; q += v * v;
  }
  atomicAdd(&sums[c], s);
  atomicAdd(&sums[128 + c], q);
}

__global__ void bn_finalize(const float* __restrict__ sums, const float* __restrict__ g,
                            const float* __restrict__ be, float* __restrict__ coef, int rows)
{
  const int c = threadIdx.x;
  const float inv_n = 1.f / (float)rows;
  const float mu = sums[c] * inv_n;
  const float var = sums[128 + c] * inv_n - mu * mu;
  const float istd = rsqrtf(var + 1e-5f);
  const float a = g[c] * istd;
  coef[c] = a;
  coef[128 + c] = be[c] - mu * a;
}

__global__ void bn_apply(const float* __restrict__ x, const float* __restrict__ resid,
                         const float* __restrict__ coef, float* __restrict__ y, long long total)
{
  const long long i = (long long)blockIdx.x * blockDim.x + threadIdx.x;
  if (i >= total) return;
  const int c = (int)(i & 127);
  float v = x[i];
  if (resid) v += resid[i];
  y[i] = v * coef[c] + coef[128 + c];
}

// ---------------------------------------------------------------------------
// Graph / segment kernels
// ---------------------------------------------------------------------------
__global__ void fill_f32(float* __restrict__ p, float v, long long n)
{
  const long long i = (long long)blockIdx.x * blockDim.x + threadIdx.x;
  if (i < n) p[i] = v;
}

// attr[eidx[e], c] += h[nidx[e], c]
__global__ void scatter_attr(const float* __restrict__ h, const int* __restrict__ nidx,
                             const int* __restrict__ eidx, float* __restrict__ attr)
{
  const long long i = (long long)blockIdx.x * blockDim.x + threadIdx.x;
  if (i >= (long long)E_INC * 128) return;
  const int e = (int)(i >> 7);
  const int c = (int)(i & 127);
  atomicAdd(&attr[(size_t)eidx[e] * 128 + c], h[(size_t)nidx[e] * 128 + c]);
}

// per-(row,head) dot of 128-wide feature with attention vector (one wave each)
__global__ void head_dot(const float* __restrict__ feat, const float* __restrict__ att,
                         float* __restrict__ out, int total, int H, int att_off)
{
  const int lane = threadIdx.x & 31;
  const int wid = blockIdx.x * (blockDim.x >> 5) + (threadIdx.x >> 5);
  if (wid >= total) return;
  const int h = wid % H;
  const float* f = feat + (size_t)wid * 128;   // feat[(n*H+h)*128]
  const float* a = att + h * 256 + att_off;
  float acc = 0.f;
#pragma unroll
  for (int c = lane; c < 128; c += 32) acc += f[c] * a[c];
#pragma unroll
  for (int off = 16; off; off >>= 1) acc += __shfl_xor(acc, off, 32);
  if (lane == 0) out[wid] = acc;
}

__global__ void alpha_max(const float* __restrict__ sx, const float* __restrict__ sa,
                          const int* __restrict__ nidx, const int* __restrict__ eidx,
                          float* __restrict__ alpha, float* __restrict__ nmax, int H)
{
  const long long i = (long long)blockIdx.x * blockDim.x + threadIdx.x;
  if (i >= (long long)E_INC * H) return;
  const int e = (int)(i / H), h = (int)(i % H);
  const float a = leaky02(sx[nidx[e] * H + h] + sa[eidx[e] * H + h]);
  alpha[i] = a;
  atomic_max_f32(&nmax[nidx[e] * H + h], a);
}

__global__ void alpha_exp(float* __restrict__ alpha, const float* __restrict__ nmax,
                          float* __restrict__ nsum, const int* __restrict__ nidx, int H)
{
  const long long i = (long long)blockIdx.x * blockDim.x + threadIdx.x;
  if (i >= (long long)E_INC * H) return;
  const int e = (int)(i / H), h = (int)(i % H);
  const float ex = expf(alpha[i] - nmax[nidx[e] * H + h]);
  alpha[i] = ex;
  atomicAdd(&nsum[nidx[e] * H + h], ex);
}

__global__ void degrees(const int* __restrict__ nidx, const int* __restrict__ eidx,
                        const float* __restrict__ w, float* __restrict__ Dd,
                        float* __restrict__ Bd)
{
  const int e = blockIdx.x * blockDim.x + threadIdx.x;
  if (e >= E_INC) return;
  atomicAdd(&Dd[nidx[e]], w[eidx[e]]);
  atomicAdd(&Bd[eidx[e]], 1.f);
}

__global__ void inv_deg(float* __restrict__ d, int n)
{
  const int i = blockIdx.x * blockDim.x + threadIdx.x;
  if (i >= n) return;
  const float v = d[i];
  d[i] = v > 0.f ? 1.f / v : 0.f;
}

__global__ void binv_fin(float* __restrict__ bd, const float* __restrict__ w, int m)
{
  const int i = blockIdx.x * blockDim.x + threadIdx.x;
  if (i >= m) return;
  const float v = bd[i];
  bd[i] = v > 0.f ? w[i] / v : 0.f;
}

// out_e[m,h,c] += softmax_alpha[e,h] * xh[n,h,c]
__global__ void msg_n2e(const float* __restrict__ xh, const float* __restrict__ alpha,
                        const float* __restrict__ nsum, const int* __restrict__ nidx,
                        const int* __restrict__ eidx, float* __restrict__ out_e, int H)
{
  const long long i = (long long)blockIdx.x * blockDim.x + threadIdx.x;
  if (i >= (long long)E_INC * H * 128) return;
  const int c = (int)(i & 127);
  const long long t = i >> 7;
  const int h = (int)(t % H);
  const int e = (int)(t / H);
  const int n = nidx[e], m = eidx[e];
  const float a = alpha[(size_t)e * H + h] / (nsum[n * H + h] + 1e-16f);
  atomicAdd(&out_e[((size_t)m * H + h) * 128 + c], a * xh[((size_t)n * H + h) * 128 + c]);
}

// out_n[n,h,c] += softmax_alpha[e,h] * out_e[m,h,c]
__global__ void msg_e2n(const float* __restrict__ out_e, const float* __restrict__ alpha,
                        const float* __restrict__ nsum, const int* __restrict__ nidx,
                        const int* __restrict__ eidx, float* __restrict__ out_n, int H)
{
  const long long i = (long long)blockIdx.x * blockDim.x + threadIdx.x;
  if (i >= (long long)E_INC * H * 128) return;
  const int c = (int)(i & 127);
  const long long t = i >> 7;
  const int h = (int)(t % H);
  const int e = (int)(t / H);
  const int n = nidx[e], m = eidx[e];
  const float a = alpha[(size_t)e * H + h] / (nsum[n * H + h] + 1e-16f);
  atomicAdd(&out_n[((size_t)n * H + h) * 128 + c], a * out_e[((size_t)m * H + h) * 128 + c]);
}

__global__ void scale_rows(float* __restrict__ x, const float* __restrict__ sc,
                           long long total, int rowlen)
{
  const long long i = (long long)blockIdx.x * blockDim.x + threadIdx.x;
  if (i >= total) return;
  x[i] *= sc[i / rowlen];
}

// res[n,c] = Dinv[n] * mean_h(out_n[n,h,c]) + bias[c]
__global__ void head_mean(const float* __restrict__ out_n, const float* __restrict__ Dinv,
                          const float* __restrict__ bias, float* __restrict__ res, int H)
{
  const long long i = (long long)blockIdx.x * blockDim.x + threadIdx.x;
  if (i >= (long long)N_NODES * 128) return;
  const int n = (int)(i >> 7), c = (int)(i & 127);
  float s = 0.f;
  for (int h = 0; h < H; ++h) s += out_n[((size_t)n * H + h) * 128 + c];
  res[i] = Dinv[n] * (s / (float)H) + bias[c];
}

// ---------------------------------------------------------------------------
// Host-side helpers
// ---------------------------------------------------------------------------
static inline int nblk(long long total, int bs) { return (int)((total + bs - 1) / bs); }

static void run_bn(const float* x, const float* resid, const float* g, const float* be,
                   float* y, float* bns, float* coef, int rows, hipStream_t s)
{
  hipMemsetAsync(bns, 0, 256 * sizeof(float), s);
  bn_stats<<<256, 128, 0, s>>>(x, resid, bns, rows);
  bn_finalize<<<1, 128, 0, s>>>(bns, g, be, coef, rows);
  const long long total = (long long)rows * 128;
  bn_apply<<<nblk(total, 256), 256, 0, s>>>(x, resid, coef, y, total);
}

static void run_hconv(const float* hin, const float* attr, const int* nidx, const int* eidx,
                      const float* lin_w, const float* att, const float* bias, int H,
                      float* xh, float* ah, float* out_e, float* out_n, float* sx, float* sa,
                      float* alpha, float* nmax, float* nsum,
                      const float* Dinv, const float* Binv, float* res, hipStream_t s)
{
  const int Nd = H * 128;
  // node / hyperedge linear projections (WMMA)
  wmma_gemm_f32<<<dim3(N_NODES / 16, Nd / 64), 128, 0, s>>>(hin, lin_w, nullptr, xh,
                                                            N_NODES, Nd, 128, 0);
  wmma_gemm_f32<<<dim3(M_HYPER / 16, Nd / 64), 128, 0, s>>>(attr, lin_w, nullptr, ah,
                                                            M_HYPER, Nd, 128, 0);
  // attention logits
  head_dot<<<nblk((long long)N_NODES * H, 8), 256, 0, s>>>(xh, att, sx, N_NODES * H, H, 0);
  head_dot<<<nblk((long long)M_HYPER * H, 8), 256, 0, s>>>(ah, att, sa, M_HYPER * H, H, 128);
  // segment softmax over destination nodes
  fill_f32<<<nblk((long long)N_NODES * H, 256), 256, 0, s>>>(nmax, -INFINITY,
                                                             (long long)N_NODES * H);
  hipMemsetAsync(nsum, 0, (size_t)N_NODES * H * sizeof(float), s);
  alpha_max<<<nblk((long long)E_INC * H, 256), 256, 0, s>>>(sx, sa, nidx, eidx, alpha, nmax, H);
  alpha_exp<<<nblk((long long)E_INC * H, 256), 256, 0, s>>>(alpha, nmax, nsum, nidx, H);
  // node -> hyperedge
  hipMemsetAsync(out_e, 0, (size_t)M_HYPER * Nd * sizeof(float), s);
  msg_n2e<<<nblk((long long)E_INC * H * 128, 256), 256, 0, s>>>(xh, alpha, nsum, nidx, eidx,
                                                                out_e, H);
  scale_rows<<<nblk((long long)M_HYPER * Nd, 256), 256, 0, s>>>(out_e, Binv,
                                                                (long long)M_HYPER * Nd, Nd);
  // hyperedge -> node  (out_n aliases xh; xh is dead by now)
  hipMemsetAsync(out_n, 0, (size_t)N_NODES * Nd * sizeof(float), s);
  msg_e2n<<<nblk((long long)E_INC * H * 128, 256), 256, 0, s>>>(out_e, alpha, nsum, nidx, eidx,
                                                                out_n, H);
  head_mean<<<nblk((long long)N_NODES * 128, 256), 256, 0, s>>>(out_n, Dinv, bias, res, H);
}

extern "C" void kernel_launch(void* const* d_in, const int* in_sizes, int n_in,
                              void* d_out, int out_size, void* d_ws, size_t ws_size,
                              hipStream_t stream)
{
  (void)in_sizes; (void)n_in; (void)out_size; (void)ws_size;
  const float* x     = (const float*)d_in[0];
  const int*   nidx  = (const int*)d_in[1];
  const int*   eidx  = (const int*)d_in[2];
  const float* w     = (const float*)d_in[3];
  const float* W1    = (const float*)d_in[4];
  const float* b1    = (const float*)d_in[5];
  const float* g1    = (const float*)d_in[6];
  const float* be1   = (const float*)d_in[7];
  const float* linw1 = (const float*)d_in[8];
  const float* att1  = (const float*)d_in[9];
  const float* bias1 = (const float*)d_in[10];
  const float* g2    = (const float*)d_in[11];
  const float* be2   = (const float*)d_in[12];
  const float* linw2 = (const float*)d_in[13];
  const float* att2  = (const float*)d_in[14];
  const float* bias2 = (const float*)d_in[15];
  const float* g3    = (const float*)d_in[16];
  const float* be3   = (const float*)d_in[17];
  const float* W2    = (const float*)d_in[18];
  const float* b2    = (const float*)d_in[19];
  const float* g4    = (const float*)d_in[20];
  const float* be4   = (const float*)d_in[21];
  float* out = (float*)d_out;

  // ---- workspace arena (floats) ----
  float* W = (float*)d_ws;
  size_t off = 0;
  auto alloc = [&](size_t n) { float* p = W + off; off += n; return p; };
  float* h1    = alloc((size_t)N_NODES * 128);
  float* h2    = alloc((size_t)N_NODES * 128);
  float* res   = alloc((size_t)N_NODES * 128);
  float* attr  = alloc((size_t)M_HYPER * 128);
  float* xh    = alloc((size_t)N_NODES * 512);   // out_n aliases this
  float* ah    = alloc((size_t)M_HYPER * 512);
  float* out_e = alloc((size_t)M_HYPER * 512);
  float* alpha = alloc((size_t)E_INC * 4);
  float* sx    = alloc((size_t)N_NODES * 4);
  float* sa    = alloc((size_t)M_HYPER * 4);
  float* nmax  = alloc((size_t)N_NODES * 4);
  float* nsum  = alloc((size_t)N_NODES * 4);
  float* Dinv  = alloc((size_t)N_NODES);
  float* Binv  = alloc((size_t)M_HYPER);
  float* bns   = alloc(256);
  float* coef  = alloc(256);
  float* out_n = xh;  // alias (xh dead once messages start using out_e)

  // 1) h1 = BN(leaky(x @ W1 + b1))
  wmma_gemm_f32<<<dim3(N_NODES / 16, 2), 128, 0, stream>>>(x, W1, b1, res,
                                                           N_NODES, 128, 128, 1);
  run_bn(res, nullptr, g1, be1, h1, bns, coef, N_NODES, stream);

  // degrees (identical for both conv layers)
  hipMemsetAsync(Dinv, 0, (size_t)N_NODES * sizeof(float), stream);
  hipMemsetAsync(Binv, 0, (size_t)M_HYPER * sizeof(float), stream);
  degrees<<<nblk(E_INC, 256), 256, 0, stream>>>(nidx, eidx, w, Dinv, Binv);
  inv_deg<<<nblk(N_NODES, 256), 256, 0, stream>>>(Dinv, N_NODES);
  binv_fin<<<nblk(M_HYPER, 256), 256, 0, stream>>>(Binv, w, M_HYPER);

  // 2) conv layer 1 (4 heads)
  hipMemsetAsync(attr, 0, (size_t)M_HYPER * 128 * sizeof(float), stream);
  scatter_attr<<<nblk((long long)E_INC * 128, 256), 256, 0, stream>>>(h1, nidx, eidx, attr);
  run_hconv(h1, attr, nidx, eidx, linw1, att1, bias1, 4,
            xh, ah, out_e, out_n, sx, sa, alpha, nmax, nsum, Dinv, Binv, res, stream);
  run_bn(h1, res, g2, be2, h2, bns, coef, N_NODES, stream);

  // 3) conv layer 2 (1 head)
  hipMemsetAsync(attr, 0, (size_t)M_HYPER * 128 * sizeof(float), stream);
  scatter_attr<<<nblk((long long)E_INC * 128, 256), 256, 0, stream>>>(h2, nidx, eidx, attr);
  run_hconv(h2, attr, nidx, eidx, linw2, att2, bias2, 1,
            xh, ah, out_e, out_n, sx, sa, alpha, nmax, nsum, Dinv, Binv, res, stream);
  run_bn(h2, res, g3, be3, h1 /* reuse as h3 */, bns, coef, N_NODES, stream);

  // 4) o = leaky(h3 @ W2 + b2);  out = BN(x + o)
  wmma_gemm_f32<<<dim3(N_NODES / 16, 2), 128, 0, stream>>>(h1, W2, b2, res,
                                                           N_NODES, 128, 128, 1);
  run_bn(x, res, g4, be4, out, bns, coef, N_NODES, stream);
}